// GCNJK_14370960572525
// MI455X (gfx1250) — compile-verified
//
#include <hip/hip_runtime.h>
#include <cstdint>

typedef float v2f __attribute__((ext_vector_type(2)));
typedef float v8f __attribute__((ext_vector_type(8)));

#define BN_EPS 1e-5f

// ---------------- degree kernels ----------------
__global__ void k_deg_init(float* __restrict__ deg, int n) {
    int i = blockIdx.x * blockDim.x + threadIdx.x;
    if (i < n) deg[i] = 1.0f;                       // self loop
}

__global__ void k_deg_count(const long long* __restrict__ dst,
                            float* __restrict__ deg, int e) {
    int i = blockIdx.x * blockDim.x + threadIdx.x;
    if (i < e) unsafeAtomicAdd(&deg[dst[i]], 1.0f); // global_atomic_add_f32
}

__global__ void k_deg_rsqrt(float* __restrict__ deg, int n) {
    int i = blockIdx.x * blockDim.x + threadIdx.x;
    if (i < n) deg[i] = rsqrtf(deg[i]);             // deg>0 always (self loop)
}

// ---------------- f32 WMMA GEMM + fused self-loop epilogue ----------------
// C[n x 128] = X[n x 128] @ W[128 x 128]; agg = C * dinv^2 (initializes accumulator buf)
// block = 256 threads = 8 waves; block covers 16 rows x 128 cols; wave w -> cols [16w,16w+16)
__global__ __launch_bounds__(256) void k_gemm128_fused(const float* __restrict__ X,
                                                       const float* __restrict__ W,
                                                       const float* __restrict__ dinv,
                                                       float* __restrict__ C,
                                                       float* __restrict__ agg,
                                                       int nrows) {
    __shared__ __align__(16) float As[16 * 132];    // 132 % 64 banks = 4 -> conflict-free rows
    const int row0 = blockIdx.x * 16;
    const int tid  = threadIdx.x;

    // async stage of X slab (16 x 128) straight into LDS (ASYNCcnt-tracked, no VGPR round-trip).
    // Tail rows are clamped: slab row r only contributes to output row r, masked at store.
    #pragma unroll
    for (int i = tid; i < 512; i += 256) {
        int rl = i >> 5, cq = i & 31;
        int r = row0 + rl;
        if (r > nrows - 1) r = nrows - 1;
        unsigned ldsOff = (unsigned)(uintptr_t)(As + rl * 132 + cq * 4);
        int voff = (r * 128 + cq * 4) * 4;          // byte offset from X (max ~51 MB, fits i32)
        asm volatile("global_load_async_to_lds_b128 %0, %1, %2"
                     :: "v"(ldsOff), "v"(voff), "s"(X) : "memory");
    }
    asm volatile("s_wait_asynccnt 0x0" ::: "memory");
    __syncthreads();

    const int wave  = tid >> 5;
    const int lane  = tid & 31;
    const int ncol  = wave * 16;
    const int mrow  = lane & 15;
    const int kh    = (lane >> 4) << 1;             // 0 or 2 (A/B K-half per ISA layout)
    const int nlane = ncol + (lane & 15);

    v8f acc = {};
    #pragma unroll
    for (int k = 0; k < 128; k += 4) {
        v2f a, b;
        a.x = As[mrow * 132 + k + kh];
        a.y = As[mrow * 132 + k + kh + 1];
        b.x = W[(size_t)(k + kh)     * 128 + nlane];
        b.y = W[(size_t)(k + kh + 1) * 128 + nlane];
        acc = __builtin_amdgcn_wmma_f32_16x16x4_f32(false, a, false, b,
                                                    (short)0, acc, false, false);
    }

    const int mo = (lane >> 4) * 8;                 // D rows: vgpr v + 8*(lane>>4)
    #pragma unroll
    for (int v = 0; v < 8; ++v) {
        int r = row0 + mo + v;
        if (r < nrows) {
            float val = acc[v];
            C[(size_t)r * 128 + nlane] = val;
            float w = dinv[r];
            agg[(size_t)r * 128 + nlane] = val * w * w;   // fused self-loop term
        }
    }
}

// ---------------- edge scatter: one wave32 per edge, float4 gather + 4 fp32 atomics ----------------
__global__ __launch_bounds__(256) void k_edge_scatter(const long long* __restrict__ src,
                                                      const long long* __restrict__ dst,
                                                      const float* __restrict__ dinv,
                                                      const float* __restrict__ h,
                                                      float* __restrict__ agg, long long e) {
    long long wid = ((long long)blockIdx.x * blockDim.x + threadIdx.x) >> 5;
    if (wid >= e) return;
    int lane = threadIdx.x & 31;
    // indices < N fit in 32 bits; readfirstlane -> SGPR bases for gather/scatter addressing
    int s = __builtin_amdgcn_readfirstlane((int)src[wid]);
    int d = __builtin_amdgcn_readfirstlane((int)dst[wid]);
    float norm = dinv[s] * dinv[d];
    float4 v = ((const float4*)(h + (size_t)s * 128))[lane];   // 512B/wave, coalesced
    float* ap = agg + (size_t)d * 128 + lane * 4;
    unsafeAtomicAdd(ap + 0, v.x * norm);
    unsafeAtomicAdd(ap + 1, v.y * norm);
    unsafeAtomicAdd(ap + 2, v.z * norm);
    unsafeAtomicAdd(ap + 3, v.w * norm);
}

// ---------------- bias + BN(eval) + ReLU, in place, float4-vectorized ----------------
__global__ void k_bn_relu4(float4* __restrict__ hbuf, const float4* __restrict__ b1,
                           const float4* __restrict__ gamma, const float4* __restrict__ beta,
                           const float4* __restrict__ mean, const float4* __restrict__ var,
                           long long total4) {
    long long i = (long long)blockIdx.x * blockDim.x + threadIdx.x;
    if (i < total4) {
        int f = (int)(i & 31);                      // 32 float4 per 128-wide row
        float4 v = hbuf[i];
        float4 B = b1[f], G = gamma[f], Bt = beta[f], M = mean[f], V = var[f];
        v.x = fmaxf((v.x + B.x - M.x) * rsqrtf(V.x + BN_EPS) * G.x + Bt.x, 0.f);
        v.y = fmaxf((v.y + B.y - M.y) * rsqrtf(V.y + BN_EPS) * G.y + Bt.y, 0.f);
        v.z = fmaxf((v.z + B.z - M.z) * rsqrtf(V.z + BN_EPS) * G.z + Bt.z, 0.f);
        v.w = fmaxf((v.w + B.w - M.w) * rsqrtf(V.w + BN_EPS) * G.w + Bt.w, 0.f);
        hbuf[i] = v;
    }
}

// ---------------- JumpingKnowledge max: jk = max(h1, agg2 + b2), float4-vectorized ----------------
__global__ void k_jkmax4(const float4* __restrict__ h1, const float4* __restrict__ agg2,
                         const float4* __restrict__ b2, float4* __restrict__ jk,
                         long long total4) {
    long long i = (long long)blockIdx.x * blockDim.x + threadIdx.x;
    if (i < total4) {
        int f = (int)(i & 31);
        float4 a = h1[i], g = agg2[i], B = b2[f];
        float4 o;
        o.x = fmaxf(a.x, g.x + B.x);
        o.y = fmaxf(a.y, g.y + B.y);
        o.z = fmaxf(a.z, g.z + B.z);
        o.w = fmaxf(a.w, g.w + B.w);
        jk[i] = o;
    }
}

// ---------------- pad Wf (128x47 -> 128x48) and bf (47 -> 48) ----------------
__global__ void k_pack_wf(const float* __restrict__ Wf, const float* __restrict__ bf,
                          float* __restrict__ Wfp, float* __restrict__ bfp) {
    int i = blockIdx.x * blockDim.x + threadIdx.x;
    if (i < 128 * 48) {
        int r = i / 48, c = i % 48;
        Wfp[i] = (c < 47) ? Wf[r * 47 + c] : 0.f;
        if (i < 48) bfp[i] = (i < 47) ? bf[i] : 0.f;
    }
}

// ---------------- final GEMM (128 -> 48) + fused log_softmax over 47 ----------------
// block = 96 threads = 3 waves; covers 16 rows x 48 cols
__global__ __launch_bounds__(96) void k_gemm_final_lsm(const float* __restrict__ JK,
                                                       const float* __restrict__ Wp,
                                                       const float* __restrict__ bp,
                                                       float* __restrict__ out, int nrows) {
    __shared__ __align__(16) float As[16 * 132];
    __shared__ float Ls[16 * 48];
    const int row0 = blockIdx.x * 16;
    const int tid  = threadIdx.x;

    for (int i = tid; i < 512; i += 96) {
        int rl = i >> 5, cq = i & 31;
        int r = row0 + rl;
        float4 v = make_float4(0.f, 0.f, 0.f, 0.f);
        if (r < nrows) v = ((const float4*)(JK + (size_t)r * 128))[cq];
        ((float4*)(As + rl * 132))[cq] = v;
    }
    __syncthreads();

    const int wave  = tid >> 5;
    const int lane  = tid & 31;
    const int ncol  = wave * 16;
    const int mrow  = lane & 15;
    const int kh    = (lane >> 4) << 1;
    const int nlane = ncol + (lane & 15);

    float bias = bp[nlane];
    v8f acc;
    #pragma unroll
    for (int v = 0; v < 8; ++v) acc[v] = bias;      // fold +bf into accumulator

    #pragma unroll
    for (int k = 0; k < 128; k += 4) {
        v2f a, b;
        a.x = As[mrow * 132 + k + kh];
        a.y = As[mrow * 132 + k + kh + 1];
        b.x = Wp[(k + kh)     * 48 + nlane];
        b.y = Wp[(k + kh + 1) * 48 + nlane];
        acc = __builtin_amdgcn_wmma_f32_16x16x4_f32(false, a, false, b,
                                                    (short)0, acc, false, false);
    }

    const int mo = (lane >> 4) * 8;
    #pragma unroll
    for (int v = 0; v < 8; ++v)
        Ls[(mo + v) * 48 + nlane] = acc[v];
    __syncthreads();

    if (tid < 16 && (row0 + tid) < nrows) {
        const float* rowp = Ls + tid * 48;
        float mx = -3.402823466e+38f;
        for (int j = 0; j < 47; ++j) mx = fmaxf(mx, rowp[j]);
        float s = 0.f;
        for (int j = 0; j < 47; ++j) s += expf(rowp[j] - mx);
        float lse = mx + logf(s);
        float* op = out + (size_t)(row0 + tid) * 47;
        for (int j = 0; j < 47; ++j) op[j] = rowp[j] - lse;
    }
}

extern "C" void kernel_launch(void* const* d_in, const int* in_sizes, int n_in,
                              void* d_out, int out_size, void* d_ws, size_t ws_size,
                              hipStream_t stream) {
    const float*     x     = (const float*)d_in[0];
    const long long* ei    = (const long long*)d_in[1];   // int64 per reference
    const float*     W1    = (const float*)d_in[2];
    const float*     b1    = (const float*)d_in[3];
    const float*     gamma = (const float*)d_in[4];
    const float*     beta  = (const float*)d_in[5];
    const float*     mean  = (const float*)d_in[6];
    const float*     var   = (const float*)d_in[7];
    const float*     W2    = (const float*)d_in[8];
    const float*     b2    = (const float*)d_in[9];
    const float*     Wf    = (const float*)d_in[10];
    const float*     bf    = (const float*)d_in[11];

    const int n = in_sizes[0] / 128;
    const int e = in_sizes[1] / 2;
    const long long* srcI = ei;          // edge_index[0]
    const long long* dstI = ei + e;      // edge_index[1]

    float* ws = (float*)d_ws;
    const size_t NF = (size_t)n * 128;
    const size_t nAl = ((size_t)n + 3) & ~(size_t)3;   // keep float4 alignment downstream
    float* dinv = ws;
    float* b0   = ws + nAl;              // h -> t2 -> jk
    float* bA   = b0 + NF;               // agg1 -> h1
    float* bB   = bA + NF;               // agg2
    float* Wfp  = bB + NF;
    float* bfp  = Wfp + 128 * 48;
    float* out  = (float*)d_out;

    const int T = 256;
    const long long tot4 = (long long)n * 32;          // float4 elements
    const int gblk = (n + 15) / 16;
    const long long ethreads = (long long)e * 32;

    // degrees -> dinv
    k_deg_init <<<(n + T - 1) / T, T, 0, stream>>>(dinv, n);
    k_deg_count<<<(e + T - 1) / T, T, 0, stream>>>(dstI, dinv, e);
    k_deg_rsqrt<<<(n + T - 1) / T, T, 0, stream>>>(dinv, n);

    // layer 1: h = x@W1 (agg seeded with self-loop in epilogue); edge scatter; +b1 -> BN -> ReLU
    k_gemm128_fused<<<gblk, 256, 0, stream>>>(x, W1, dinv, b0, bA, n);
    k_edge_scatter<<<(int)((ethreads + T - 1) / T), T, 0, stream>>>(srcI, dstI, dinv, b0, bA, e);
    k_bn_relu4<<<(int)((tot4 + T - 1) / T), T, 0, stream>>>(
        (float4*)bA, (const float4*)b1, (const float4*)gamma, (const float4*)beta,
        (const float4*)mean, (const float4*)var, tot4);

    // layer 2: t2 = h1@W2 (agg2 seeded in epilogue); edge scatter
    k_gemm128_fused<<<gblk, 256, 0, stream>>>(bA, W2, dinv, b0, bB, n);
    k_edge_scatter<<<(int)((ethreads + T - 1) / T), T, 0, stream>>>(srcI, dstI, dinv, b0, bB, e);

    // jk = max(h1, agg2 + b2)
    k_jkmax4<<<(int)((tot4 + T - 1) / T), T, 0, stream>>>(
        (const float4*)bA, (const float4*)bB, (const float4*)b2, (float4*)b0, tot4);

    // final projection + log_softmax
    k_pack_wf<<<(128 * 48 + T - 1) / T, T, 0, stream>>>(Wf, bf, Wfp, bfp);
    k_gemm_final_lsm<<<gblk, 96, 0, stream>>>(b0, Wfp, bfp, out, n);
}